// Model_18554258719406
// MI455X (gfx1250) — compile-verified
//
#include <hip/hip_runtime.h>

typedef __attribute__((ext_vector_type(2))) float v2f;
typedef __attribute__((ext_vector_type(8))) float v8f;

namespace {
constexpr int NB = 8, VWS = 4, TDIM = 5, VDIM = 17, CIN = 18, CH = 16, HW = 256;
constexpr int NODES = NB * VWS * TDIM * VDIM;     // 2720 graph nodes (n,vw,t,v)
constexpr int NODE_ELEMS = CH * HW;               // 4096 floats per node (c,h,w)
constexpr float BN_EPS = 1e-5f;
}

// ---------------------------------------------------------------------------
// 1x1 conv (graphical branch): Y[node, o, p] = sum_c W[o,c] * X[node, c, p] + B[o]
// One wave per (node, 16-pixel tile). D(16x16) = A(16xK weights) x B(Kx16 pixels)
// via chained v_wmma_f32_16x16x4_f32, accumulator seeded with the bias.
// Grid is exact -> EXEC all ones at every WMMA. K-tail (K=18) is handled
// branchlessly: clamped in-bounds address + 0/1 mask multiply (v_cndmask, not
// s_and_saveexec), so loads stay clause-batched.
// ---------------------------------------------------------------------------
template <int K>
__global__ __launch_bounds__(256) void conv1_wmma(
    const float* __restrict__ X, float* __restrict__ Y,
    const float* __restrict__ W, const float* __restrict__ B) {
  const int wave = (blockIdx.x * 256 + threadIdx.x) >> 5;
  const int lane = threadIdx.x & 31;
  const int node = wave >> 4;
  const int tile = wave & 15;
  const int half = lane >> 4;  // lanes 16-31 hold the upper K pair / upper C rows
  const int lm   = lane & 15;

  v8f acc;
#pragma unroll
  for (int i = 0; i < 8; ++i) acc[i] = B[i + 8 * half];  // bias for row M=i+8*half

  const float* Xn = X + (size_t)node * K * HW + tile * 16 + lm;
  const int cb = 2 * half;
#pragma unroll
  for (int kk = 0; kk < K; kk += 4) {
    v2f a, b;
    if (kk + 3 < K) {  // constant-folded per unrolled iteration: fast path
      a.x = W[lm * K + kk + cb];
      a.y = W[lm * K + kk + cb + 1];
      b.x = Xn[(kk + cb) * HW];
      b.y = Xn[(kk + cb + 1) * HW];
    } else {           // K tail: branchless clamp + mask (stays in bounds)
      const int c0 = kk + cb, c1 = kk + cb + 1;
      const int cc0 = (c0 < K) ? c0 : (K - 1);
      const int cc1 = (c1 < K) ? c1 : (K - 1);
      const float m0 = (c0 < K) ? 1.0f : 0.0f;
      const float m1 = (c1 < K) ? 1.0f : 0.0f;
      a.x = m0 * W[lm * K + cc0];
      a.y = m1 * W[lm * K + cc1];
      b.x = m0 * Xn[cc0 * HW];
      b.y = m1 * Xn[cc1 * HW];
    }
    acc = __builtin_amdgcn_wmma_f32_16x16x4_f32(false, a, false, b, (short)0,
                                                acc, false, false);
  }

  float* Yn = Y + (size_t)node * NODE_ELEMS + tile * 16 + lm;
#pragma unroll
  for (int i = 0; i < 8; ++i) Yn[(i + 8 * half) * HW] = acc[i];
}

// ---------------------------------------------------------------------------
// tcn 1x1 conv with BN2 folded into weights/bias, + residual, accumulating the
// per-relation sum:  Acc (+)= (s2 o W) y + (s2*b + b2) + Res,  optional ReLU on
// the final relation. Compile-time flags -> straight-line epilogue, batched
// loads, no scalar branching around the stores.
// ---------------------------------------------------------------------------
template <bool HAS_RES, bool DO_ACCUM, bool DO_RELU>
__global__ __launch_bounds__(256) void conv2_wmma(
    const float* __restrict__ X, float* __restrict__ Acc,
    const float* __restrict__ W, const float* __restrict__ B,
    const float* __restrict__ G2, const float* __restrict__ B2,
    const float* __restrict__ Res) {
  const int wave = (blockIdx.x * 256 + threadIdx.x) >> 5;
  const int lane = threadIdx.x & 31;
  const int node = wave >> 4;
  const int tile = wave & 15;
  const int half = lane >> 4;
  const int lm   = lane & 15;

  const float inv = rsqrtf(1.0f + BN_EPS);
  const float sA  = G2[lm] * inv;  // BN2 scale folded into A-matrix row lm

  v8f acc;
#pragma unroll
  for (int i = 0; i < 8; ++i) {
    const int M = i + 8 * half;
    acc[i] = B[M] * (G2[M] * inv) + B2[M];  // folded bias
  }

  const float* Xn = X + (size_t)node * NODE_ELEMS + tile * 16 + lm;
  const int cb = 2 * half;
#pragma unroll
  for (int kk = 0; kk < CH; kk += 4) {
    v2f a, b;
    a.x = sA * W[lm * CH + kk + cb];
    a.y = sA * W[lm * CH + kk + cb + 1];
    b.x = Xn[(kk + cb) * HW];
    b.y = Xn[(kk + cb + 1) * HW];
    acc = __builtin_amdgcn_wmma_f32_16x16x4_f32(false, a, false, b, (short)0,
                                                acc, false, false);
  }

  const size_t base = (size_t)node * NODE_ELEMS + tile * 16 + lm;

  float rv[8], av[8];
  if constexpr (HAS_RES) {
#pragma unroll
    for (int i = 0; i < 8; ++i) rv[i] = Res[base + (size_t)(i + 8 * half) * HW];
  }
  if constexpr (DO_ACCUM) {
#pragma unroll
    for (int i = 0; i < 8; ++i) av[i] = Acc[base + (size_t)(i + 8 * half) * HW];
  }
#pragma unroll
  for (int i = 0; i < 8; ++i) {
    float v = acc[i];
    if constexpr (HAS_RES)  v += rv[i];
    if constexpr (DO_ACCUM) v += av[i];
    if constexpr (DO_RELU)  v = fmaxf(v, 0.0f);
    Acc[base + (size_t)(i + 8 * half) * HW] = v;
  }
}

// ---------------------------------------------------------------------------
// Adjacency aggregation along a strided axis + BN1 + ReLU.
// Canonical tensor viewed as outer x D x inner. Each thread owns one
// (outer, inner) site: loads its D inputs ONCE into registers, computes all D
// outputs (D^2 FMAs, adjacency reads are block-uniform scalar loads), stores D
// outputs. Cuts vector-load count by D vs. the naive one-thread-per-output map.
// ---------------------------------------------------------------------------
template <int D>
__global__ __launch_bounds__(256) void aggregate_bn_relu(
    const float* __restrict__ In, float* __restrict__ Out,
    const float* __restrict__ Adj, int innerBlocks,
    const float* __restrict__ G1, const float* __restrict__ B1) {
  const int ib    = blockIdx.x % innerBlocks;
  const int outer = blockIdx.x / innerBlocks;
  const long inner    = (long)innerBlocks * 256;
  const long innerIdx = (long)ib * 256 + threadIdx.x;

  const float* in0 = In + (long)outer * D * inner + innerIdx;
  float xin[D];
#pragma unroll
  for (int j = 0; j < D; ++j) xin[j] = in0[(long)j * inner];

  const int c = ib & 15;  // channel stride is 256 = one block (uniform)
  const float inv = rsqrtf(1.0f + BN_EPS);
  const float sc = G1[c] * inv, bo = B1[c];

  float* out0 = Out + (long)outer * D * inner + innerIdx;
#pragma unroll
  for (int u = 0; u < D; ++u) {
    float s = 0.0f;
#pragma unroll
    for (int j = 0; j < D; ++j) s = fmaf(Adj[u * D + j], xin[j], s);
    out0[(long)u * inner] = fmaxf(fmaf(s, sc, bo), 0.0f);
  }
}

// ---------------------------------------------------------------------------
extern "C" void kernel_launch(void* const* d_in, const int* in_sizes, int n_in,
                              void* d_out, int out_size, void* d_ws, size_t ws_size,
                              hipStream_t stream) {
  const float* x    = (const float*)d_in[0];
  const float* Asp  = (const float*)d_in[1];
  const float* Atm  = (const float*)d_in[2];
  const float* Avw  = (const float*)d_in[3];
  const float* Wg0  = (const float*)d_in[4];
  const float* bg0  = (const float*)d_in[5];
  const float* Wg   = (const float*)d_in[6];
  const float* bg   = (const float*)d_in[7];
  const float* bn1g = (const float*)d_in[8];
  const float* bn1b = (const float*)d_in[9];
  const float* Wt   = (const float*)d_in[10];
  const float* bt   = (const float*)d_in[11];
  const float* bn2g = (const float*)d_in[12];
  const float* bn2b = (const float*)d_in[13];

  float* ws = (float*)d_ws;
  const size_t SZ = (size_t)NODES * NODE_ELEMS;  // 11,141,120 floats
  float* lay0 = ws;            // layer0 out -> layer1 in
  float* lay1 = ws + SZ;       // layer1 out -> layer2 in
  float* tmp1 = ws + 2 * SZ;   // conv1 result
  float* tmp2 = ws + 3 * SZ;   // aggregated+BN+ReLU result
  float* outp = (float*)d_out;

  const int convBlocks = (NODES * 16) / 8;  // one wave per (node, pixel tile)

  for (int layer = 0; layer < 3; ++layer) {
    const float* in  = (layer == 0) ? x : (layer == 1 ? lay0 : lay1);
    float*       acc = (layer == 0) ? lay0 : (layer == 1 ? lay1 : outp);

    for (int k = 0; k < 7; ++k) {
      // 1) HeatmapGraphical 1x1 conv (channel GEMM via f32 WMMA)
      if (layer == 0) {
        conv1_wmma<CIN><<<convBlocks, 256, 0, stream>>>(
            in, tmp1, Wg0 + k * CH * CIN, bg0 + k * CH);
      } else {
        conv1_wmma<CH><<<convBlocks, 256, 0, stream>>>(
            in, tmp1, Wg + ((layer - 1) * 7 + k) * CH * CH,
            bg + ((layer - 1) * 7 + k) * CH);
      }

      // 2) adjacency aggregation over the relation's graph axis + BN1 + ReLU
      const int lk = layer * 7 + k;
      if (k < 4) {
        const int innerBlocks = NODE_ELEMS / 256;                // 16
        aggregate_bn_relu<VDIM><<<(NB * VWS * TDIM) * innerBlocks, 256, 0, stream>>>(
            tmp1, tmp2, Asp + k * VDIM * VDIM, innerBlocks,
            bn1g + lk * CH, bn1b + lk * CH);
      } else if (k < 6) {
        const int innerBlocks = VDIM * NODE_ELEMS / 256;         // 272
        aggregate_bn_relu<TDIM><<<(NB * VWS) * innerBlocks, 256, 0, stream>>>(
            tmp1, tmp2, Atm + (k - 4) * TDIM * TDIM, innerBlocks,
            bn1g + lk * CH, bn1b + lk * CH);
      } else {
        const int innerBlocks = TDIM * VDIM * NODE_ELEMS / 256;  // 1360
        aggregate_bn_relu<VWS><<<NB * innerBlocks, 256, 0, stream>>>(
            tmp1, tmp2, Avw, innerBlocks,
            bn1g + lk * CH, bn1b + lk * CH);
      }

      // 3) tcn 1x1 conv + folded BN2 + residual, accumulated over relations;
      //    final relation also applies the layer-output ReLU.
      const float* Wrow  = Wt + lk * CH * CH;
      const float* brow  = bt + lk * CH;
      const float* grow  = bn2g + lk * CH;
      const float* brow2 = bn2b + lk * CH;
      if (layer == 0) {
        if (k == 0)
          conv2_wmma<false, false, false><<<convBlocks, 256, 0, stream>>>(
              tmp2, acc, Wrow, brow, grow, brow2, nullptr);
        else if (k < 6)
          conv2_wmma<false, true, false><<<convBlocks, 256, 0, stream>>>(
              tmp2, acc, Wrow, brow, grow, brow2, nullptr);
        else
          conv2_wmma<false, true, true><<<convBlocks, 256, 0, stream>>>(
              tmp2, acc, Wrow, brow, grow, brow2, nullptr);
      } else {
        if (k == 0)
          conv2_wmma<true, false, false><<<convBlocks, 256, 0, stream>>>(
              tmp2, acc, Wrow, brow, grow, brow2, in);
        else if (k < 6)
          conv2_wmma<true, true, false><<<convBlocks, 256, 0, stream>>>(
              tmp2, acc, Wrow, brow, grow, brow2, in);
        else
          conv2_wmma<true, true, true><<<convBlocks, 256, 0, stream>>>(
              tmp2, acc, Wrow, brow, grow, brow2, in);
      }
    }
  }
  (void)in_sizes; (void)n_in; (void)out_size; (void)ws_size;
}